// HiddenLayer_20126216749058
// MI455X (gfx1250) — compile-verified
//
#include <hip/hip_runtime.h>
#include <hip/hip_bf16.h>
#include <math.h>

// ---------------- problem constants ----------------
#define Nn 1024
#define Qq 32
#define Mm 256
#define Dd 32
#define JITTER 1e-6f

typedef __attribute__((ext_vector_type(2))) float v2f;
typedef __attribute__((ext_vector_type(8))) float v8f;

// FP32 WMMA 16x16 tile, K=4 (V_WMMA_F32_16X16X4_F32).
__device__ inline v8f wmma4(v2f a, v2f b, v8f c) {
#if __has_builtin(__builtin_amdgcn_wmma_f32_16x16x4_f32)
  return __builtin_amdgcn_wmma_f32_16x16x4_f32(false, a, false, b, (short)0, c,
                                               false, false);
#else
  // Fallback so the rest of the pipeline still compiles (histogram -> wmma=0)
  c[0] += a[0] * b[0] + a[1] * b[1];
  return c;
#endif
}

// ---------------- workspace layout (floats) ----------------
constexpr size_t NQ = (size_t)Nn * Qq;
constexpr size_t NM = (size_t)Nn * Mm;
constexpr size_t MM = (size_t)Mm * Mm;
constexpr size_t OFF_W1   = 0;
constexpr size_t OFF_MW1  = OFF_W1 + NQ;
constexpr size_t OFF_W2   = OFF_MW1 + NQ;
constexpr size_t OFF_MW2  = OFF_W2 + NQ;
constexpr size_t OFF_C1   = OFF_MW2 + NQ;
constexpr size_t OFF_G2   = OFF_C1 + Nn;
constexpr size_t OFF_PSI1 = OFF_G2 + Nn;
constexpr size_t OFF_E2   = OFF_PSI1 + NM;
constexpr size_t OFF_QMK  = OFF_E2 + NM;
constexpr size_t OFF_L    = OFF_QMK + MM;
constexpr size_t OFF_P2S  = OFF_L + MM;
constexpr size_t OFF_TMM  = OFF_P2S + MM;
constexpr size_t OFF_KP2  = OFF_TMM + MM;
constexpr size_t OFF_TRM  = OFF_KP2 + MM;
constexpr size_t OFF_KP2K = OFF_TRM + MM;
constexpr size_t OFF_UUT  = OFF_KP2K + MM;
constexpr size_t OFF_CSUM = OFF_UUT + MM;
constexpr size_t OFF_WM   = OFF_CSUM + MM;
constexpr size_t OFF_P1T  = OFF_WM + MM;
constexpr size_t OFF_YBIG = OFF_P1T + NM;
constexpr size_t OFF_PT   = OFF_YBIG + NM;
constexpr size_t OFF_H    = OFF_PT + NM;
constexpr size_t OFF_ATMP = OFF_H + NM;
constexpr size_t OFF_ALPH = OFF_ATMP + (size_t)Mm * Dd;
constexpr size_t OFF_TC   = OFF_ALPH + (size_t)Mm * Dd;   // 128*32*256 chunk

// output layout: forward_mean (N*D), forward_var (N*D*D), lml (1)
constexpr size_t FM_OFF  = 0;
constexpr size_t FV_OFF  = (size_t)Nn * Dd;
constexpr size_t LML_OFF = FV_OFF + (size_t)Nn * Dd * Dd;

// ---------------- per-n statistics ----------------
__global__ void stats_kernel(const float* __restrict__ Xm,
                             const float* __restrict__ Xv,
                             const float* __restrict__ ls,
                             const float* __restrict__ varp,
                             float* __restrict__ w1, float* __restrict__ mw1,
                             float* __restrict__ w2, float* __restrict__ mw2,
                             float* __restrict__ c1, float* __restrict__ g2) {
  int n = blockIdx.x * blockDim.x + threadIdx.x;
  if (n >= Nn) return;
  float ld1 = 0.f, A1 = 0.f, ld2 = 0.f, A2 = 0.f;
  for (int q = 0; q < Qq; ++q) {
    float l = ls[q];
    float l2 = l * l;
    float mu = Xm[n * Qq + q];
    float S  = Xv[n * Qq + q];
    float d1 = l2 + S;        float iw1 = 1.0f / d1;
    float d2 = l2 + 2.0f * S; float iw2 = 1.0f / d2;
    w1[n * Qq + q]  = iw1;  mw1[n * Qq + q] = mu * iw1;
    w2[n * Qq + q]  = iw2;  mw2[n * Qq + q] = mu * iw2;
    ld1 += logf(d1 / l2);  A1 += mu * mu * iw1;
    ld2 += logf(d2 / l2);  A2 += mu * mu * iw2;
  }
  c1[n] = -0.5f * (ld1 + A1);
  g2[n] = 2.0f * logf(varp[0]) - 0.5f * ld2 - A2;
}

// ---------------- psi1 and E2 = B2 - 0.25*C2 ----------------
__global__ void psi1e2_kernel(const float* __restrict__ Z,
                              const float* __restrict__ varp,
                              const float* __restrict__ w1,
                              const float* __restrict__ mw1,
                              const float* __restrict__ w2,
                              const float* __restrict__ mw2,
                              const float* __restrict__ c1,
                              float* __restrict__ psi1, float* __restrict__ E2) {
  int t = blockIdx.x * blockDim.x + threadIdx.x;
  if (t >= Nn * Mm) return;
  int n = t >> 8, m = t & (Mm - 1);
  float s1 = 0.f, s2 = 0.f;
  for (int q = 0; q < Qq; ++q) {
    float z = Z[m * Qq + q];
    s1 += z * (mw1[n * Qq + q] - 0.5f * w1[n * Qq + q] * z);
    s2 += z * (mw2[n * Qq + q] - 0.25f * w2[n * Qq + q] * z);
  }
  psi1[t] = varp[0] * expf(c1[n] + s1);
  E2[t] = s2;
}

// ---------------- qmk and Kmm (into L buffer) ----------------
__global__ void kmm_kernel(const float* __restrict__ Z,
                           const float* __restrict__ ls,
                           const float* __restrict__ varp,
                           float* __restrict__ qmk, float* __restrict__ L) {
  int t = blockIdx.x * blockDim.x + threadIdx.x;
  if (t >= Mm * Mm) return;
  int m = t >> 8, k = t & (Mm - 1);
  float s = 0.f;
  for (int q = 0; q < Qq; ++q) {
    float d = (Z[m * Qq + q] - Z[k * Qq + q]) / ls[q];
    s += d * d;
  }
  qmk[t] = s;
  L[t] = varp[0] * expf(-0.5f * s) + ((m == k) ? JITTER : 0.0f);
}

// ---------------- in-place Cholesky, one workgroup ----------------
__global__ __launch_bounds__(256) void chol_kernel(float* __restrict__ L) {
  int t = threadIdx.x;
  for (int j = 0; j < Mm; ++j) {
    if (t == 0) L[j * Mm + j] = sqrtf(L[j * Mm + j]);
    __threadfence_block();
    __syncthreads();
    float ljj = L[j * Mm + j];
    int i = j + 1 + t;
    if (i < Mm) L[i * Mm + j] /= ljj;
    __threadfence_block();
    __syncthreads();
    if (i < Mm) {
      float lij = L[i * Mm + j];
      for (int k = j + 1; k <= i; ++k) L[i * Mm + k] -= lij * L[k * Mm + j];
    }
    __threadfence_block();
    __syncthreads();
  }
}

// ---------------- triangular solves (column per thread) ----------------
__global__ void trsm_fwd(const float* __restrict__ L, const float* __restrict__ B,
                         float* __restrict__ Y, int ncols) {
  int c = blockIdx.x * blockDim.x + threadIdx.x;
  if (c >= ncols) return;
  for (int i = 0; i < Mm; ++i) {
    float s = B[i * ncols + c];
    for (int k = 0; k < i; ++k) s -= L[i * Mm + k] * Y[k * ncols + c];
    Y[i * ncols + c] = s / L[i * Mm + i];
  }
}

__global__ void trsm_bwd(const float* __restrict__ L, const float* __restrict__ B,
                         float* __restrict__ Y, int ncols) {
  int c = blockIdx.x * blockDim.x + threadIdx.x;
  if (c >= ncols) return;
  for (int i = Mm - 1; i >= 0; --i) {
    float s = B[i * ncols + c];
    for (int k = i + 1; k < Mm; ++k) s -= L[k * Mm + i] * Y[k * ncols + c];
    Y[i * ncols + c] = s / L[i * Mm + i];
  }
}

__global__ void transpose_kernel(const float* __restrict__ A,
                                 float* __restrict__ B, int rows, int cols) {
  int t = blockIdx.x * blockDim.x + threadIdx.x;
  if (t >= rows * cols) return;
  int r = t / cols, c = t % cols;
  B[c * rows + r] = A[t];
}

// ---------------- psi2_sum via FP32 WMMA (D2 contraction) ----------------
// One wave per (m, 16-wide k-tile). D2 tile = w2(16n x Q) @ zmk(Q x 16k).
__global__ __launch_bounds__(32) void psi2_kernel(const float* __restrict__ Z,
                                                  const float* __restrict__ w2,
                                                  const float* __restrict__ g2,
                                                  const float* __restrict__ E2,
                                                  const float* __restrict__ qmk,
                                                  float* __restrict__ psi2s) {
  int lane = threadIdx.x;
  int bid = blockIdx.x;            // 4096 = 256 m * 16 k-tiles
  int m = bid >> 4;
  int k0 = (bid & 15) << 4;
  int col = lane & 15;             // A-row and B/D-col for this lane
  int kb = (lane >> 4) << 1;       // K pair: 0,1 or 2,3
  int rowbase = (lane >> 4) << 3;  // D rows 0..7 or 8..15
  int kcol = k0 + col;
  float acc = 0.0f;
  for (int n0 = 0; n0 < Nn; n0 += 16) {
    v8f c = {};
    for (int q0 = 0; q0 < Qq; q0 += 4) {
      v2f a, b;
      a.x = w2[(n0 + col) * Qq + q0 + kb];
      a.y = w2[(n0 + col) * Qq + q0 + kb + 1];
      b.x = Z[m * Qq + q0 + kb]     * Z[kcol * Qq + q0 + kb];
      b.y = Z[m * Qq + q0 + kb + 1] * Z[kcol * Qq + q0 + kb + 1];
      c = wmma4(a, b, c);
    }
#pragma unroll
    for (int r = 0; r < 8; ++r) {
      int nn = n0 + rowbase + r;
      acc += expf(g2[nn] + E2[nn * Mm + m] + E2[nn * Mm + kcol] - 0.5f * c[r]);
    }
  }
  acc += __shfl_xor(acc, 16, 32);
  if (lane < 16) {
    psi2s[m * Mm + kcol] = expf(-0.25f * qmk[m * Mm + kcol]) * acc;
  }
}

// ---------------- tmp[n,i,l] = sum_j P[n,j] * q_chol[i,j,l], chunked over n ----
__global__ __launch_bounds__(32) void tmul_kernel(const float* __restrict__ PT,
                                                  const float* __restrict__ qs,
                                                  float* __restrict__ Tc, int n0c) {
  int lane = threadIdx.x;
  int bid = blockIdx.x;            // 4096 = 32 d * 8 n-tiles * 16 l-tiles
  int d = bid >> 7;
  int nt = (bid >> 4) & 7;
  int lt = bid & 15;
  int col = lane & 15;
  int kb = (lane >> 4) << 1;
  int rowbase = (lane >> 4) << 3;
  int l = lt * 16 + col;
  int nglob = n0c + nt * 16 + col;  // A-row index
  v8f c = {};
  for (int j0 = 0; j0 < Mm; j0 += 4) {
    int j = j0 + kb;
    v2f a, b;
    a.x = PT[j * Nn + nglob];
    a.y = PT[(j + 1) * Nn + nglob];
    b.x = (j >= l)     ? qs[((size_t)j * Mm + l) * Dd + d] : 0.0f;
    b.y = (j + 1 >= l) ? qs[((size_t)(j + 1) * Mm + l) * Dd + d] : 0.0f;
    c = wmma4(a, b, c);
  }
#pragma unroll
  for (int r = 0; r < 8; ++r) {
    int nnl = nt * 16 + rowbase + r;  // local n within chunk (0..127)
    Tc[((size_t)nnl * Dd + d) * Mm + lt * 16 + col] = c[r];
  }
}

// ---------------- forward_var[n] = T[n] T[n]^T + I/beta (WMMA Gram) --------
__global__ __launch_bounds__(32) void gram_kernel(const float* __restrict__ Tc,
                                                  const float* __restrict__ betap,
                                                  float* __restrict__ out_fv,
                                                  int n0c) {
  int lane = threadIdx.x;
  int bid = blockIdx.x;            // 512 = 128 nn * 2 * 2 tiles
  int nn = bid >> 2;
  int it = (bid >> 1) & 1;
  int kt = bid & 1;
  int col = lane & 15;
  int kb = (lane >> 4) << 1;
  int rowbase = (lane >> 4) << 3;
  int i_row = it * 16 + col;
  int k_col = kt * 16 + col;
  const float* Tn = Tc + (size_t)nn * Dd * Mm;
  v8f c = {};
  for (int l0 = 0; l0 < Mm; l0 += 4) {
    v2f a, b;
    a.x = Tn[i_row * Mm + l0 + kb];
    a.y = Tn[i_row * Mm + l0 + kb + 1];
    b.x = Tn[k_col * Mm + l0 + kb];
    b.y = Tn[k_col * Mm + l0 + kb + 1];
    c = wmma4(a, b, c);
  }
  float invb = 1.0f / betap[0];
  int n = n0c + nn;
#pragma unroll
  for (int r = 0; r < 8; ++r) {
    int i = it * 16 + rowbase + r;
    out_fv[((size_t)n * Dd + i) * Dd + k_col] =
        c[r] + ((i == k_col) ? invb : 0.0f);
  }
}

// ---------------- forward_mean ----------------
__global__ void fmean_kernel(const float* __restrict__ PT,
                             const float* __restrict__ qmu,
                             float* __restrict__ out) {
  int t = blockIdx.x * blockDim.x + threadIdx.x;
  if (t >= Nn * Dd) return;
  int n = t >> 5, d = t & 31;
  float s = 0.f;
  for (int j = 0; j < Mm; ++j) s += PT[j * Nn + n] * qmu[j * Dd + d];
  out[t] = s;
}

// ---------------- uuT and Csum = sum_d q_cov_d ----------------
__global__ void uut_kernel(const float* __restrict__ qmu,
                           const float* __restrict__ qs,
                           float* __restrict__ uuT, float* __restrict__ Csum) {
  int t = blockIdx.x * blockDim.x + threadIdx.x;
  if (t >= Mm * Mm) return;
  int a = t >> 8, b = t & (Mm - 1);
  float mmv = 0.f;
  for (int d = 0; d < Dd; ++d) mmv += qmu[a * Dd + d] * qmu[b * Dd + d];
  int lim = (a < b) ? a : b;
  float cs = 0.f;
  for (int l = 0; l <= lim; ++l) {
    const float* pa = qs + ((size_t)a * Mm + l) * Dd;
    const float* pb = qs + ((size_t)b * Mm + l) * Dd;
    for (int d = 0; d < Dd; ++d) cs += pa[d] * pb[d];
  }
  uuT[t] = mmv + cs;
  Csum[t] = cs;
}

// ---------------- H = P @ uuT ----------------
__global__ void h_kernel(const float* __restrict__ PT,
                         const float* __restrict__ uuT, float* __restrict__ H) {
  int t = blockIdx.x * blockDim.x + threadIdx.x;
  if (t >= Nn * Mm) return;
  int n = t >> 8, a = t & (Mm - 1);
  float s = 0.f;
  for (int b = 0; b < Mm; ++b) s += PT[b * Nn + n] * uuT[a * Mm + b];
  H[t] = s;
}

// ---------------- final deterministic reductions -> lml ----------------
__global__ __launch_bounds__(256) void lml_kernel(
    const float* __restrict__ KP2, const float* __restrict__ W,
    const float* __restrict__ L, const float* __restrict__ qs,
    const float* __restrict__ qmu, const float* __restrict__ alpha,
    const float* __restrict__ KP2K, const float* __restrict__ uuT,
    const float* __restrict__ H, const float* __restrict__ PT,
    const float* __restrict__ varp, const float* __restrict__ betap,
    float* __restrict__ out_lml) {
  __shared__ float red[256];
  int t = threadIdx.x;
  auto bsum = [&](float v) -> float {
    red[t] = v;
    __syncthreads();
    for (int s = 128; s > 0; s >>= 1) {
      if (t < s) red[t] += red[t + s];
      __syncthreads();
    }
    float r = red[0];
    __syncthreads();
    return r;
  };
  float trKP2 = bsum(KP2[t * Mm + t]);
  float trW = bsum(W[t * Mm + t]);            // tr_kl
  float ldK = bsum(2.0f * logf(L[t * Mm + t]));
  float v = 0.f;
  for (int idx = t; idx < Mm * Dd; idx += 256) {
    int j = idx >> 5, d = idx & 31;
    float sq = qs[((size_t)j * Mm + j) * Dd + d];
    v += logf(sq * sq);
  }
  float ldS = bsum(v);
  v = 0.f;
  for (int idx = t; idx < Mm * Dd; idx += 256) v += qmu[idx] * alpha[idx];
  float mahal = bsum(v);
  v = 0.f;
  for (int idx = t; idx < Mm * Mm; idx += 256) v += KP2K[idx] * uuT[idx];
  float S1 = bsum(v);
  v = 0.f;
  for (int idx = t; idx < Nn * Mm; idx += 256) {
    int n = idx >> 8, a = idx & (Mm - 1);
    v += H[idx] * PT[a * Nn + n];
  }
  float S2 = bsum(v);
  if (t == 0) {
    float var = varp[0], beta = betap[0];
    float trace = (float)Nn * var - trKP2;
    float lml = -0.5f * beta * (float)Dd * trace;
    float kl =
        0.5f * (mahal + trW - (float)(Mm * Dd) + (float)Dd * ldK - ldS);
    lml -= kl;
    lml -= 0.5f * beta * (S1 - S2);
    out_lml[0] = lml;
  }
}

// ---------------- launch ----------------
extern "C" void kernel_launch(void* const* d_in, const int* in_sizes, int n_in,
                              void* d_out, int out_size, void* d_ws,
                              size_t ws_size, hipStream_t stream) {
  const float* Xm   = (const float*)d_in[0];
  const float* Xv   = (const float*)d_in[1];
  const float* Z    = (const float*)d_in[2];
  const float* qmu  = (const float*)d_in[3];
  const float* qs   = (const float*)d_in[4];
  const float* ls   = (const float*)d_in[5];
  const float* varp = (const float*)d_in[6];
  const float* betap= (const float*)d_in[7];
  float* out = (float*)d_out;
  float* ws = (float*)d_ws;

  float* w1   = ws + OFF_W1;
  float* mw1  = ws + OFF_MW1;
  float* w2   = ws + OFF_W2;
  float* mw2  = ws + OFF_MW2;
  float* c1   = ws + OFF_C1;
  float* g2   = ws + OFF_G2;
  float* psi1 = ws + OFF_PSI1;
  float* E2   = ws + OFF_E2;
  float* qmk  = ws + OFF_QMK;
  float* Lm   = ws + OFF_L;
  float* p2s  = ws + OFF_P2S;
  float* tmm  = ws + OFF_TMM;
  float* KP2  = ws + OFF_KP2;
  float* trm  = ws + OFF_TRM;
  float* KP2K = ws + OFF_KP2K;
  float* uuT  = ws + OFF_UUT;
  float* Csum = ws + OFF_CSUM;
  float* Wm   = ws + OFF_WM;
  float* p1T  = ws + OFF_P1T;
  float* Ybig = ws + OFF_YBIG;
  float* PT   = ws + OFF_PT;
  float* H    = ws + OFF_H;
  float* aTmp = ws + OFF_ATMP;
  float* alph = ws + OFF_ALPH;
  float* Tc   = ws + OFF_TC;

  // stage 1: per-n statistics and pairwise pieces
  stats_kernel<<<4, 256, 0, stream>>>(Xm, Xv, ls, varp, w1, mw1, w2, mw2, c1, g2);
  psi1e2_kernel<<<1024, 256, 0, stream>>>(Z, varp, w1, mw1, w2, mw2, c1, psi1, E2);
  kmm_kernel<<<256, 256, 0, stream>>>(Z, ls, varp, qmk, Lm);
  chol_kernel<<<1, 256, 0, stream>>>(Lm);

  // stage 2: psi1Kmmi^T = cho_solve(L, psi1^T)
  transpose_kernel<<<1024, 256, 0, stream>>>(psi1, p1T, Nn, Mm);
  trsm_fwd<<<4, 256, 0, stream>>>(Lm, p1T, Ybig, Nn);
  trsm_bwd<<<4, 256, 0, stream>>>(Lm, Ybig, PT, Nn);

  // alpha = cho_solve(L, q_mu)
  trsm_fwd<<<1, 32, 0, stream>>>(Lm, qmu, aTmp, Dd);
  trsm_bwd<<<1, 32, 0, stream>>>(Lm, aTmp, alph, Dd);

  // stage 3: psi2_sum (WMMA) and K^{-1} applications
  psi2_kernel<<<4096, 32, 0, stream>>>(Z, w2, g2, E2, qmk, p2s);
  trsm_fwd<<<1, 256, 0, stream>>>(Lm, p2s, tmm, Mm);
  trsm_bwd<<<1, 256, 0, stream>>>(Lm, tmm, KP2, Mm);        // KmmiPsi2
  transpose_kernel<<<256, 256, 0, stream>>>(KP2, trm, Mm, Mm);
  trsm_fwd<<<1, 256, 0, stream>>>(Lm, trm, tmm, Mm);
  trsm_bwd<<<1, 256, 0, stream>>>(Lm, tmm, KP2K, Mm);       // KmmiPsi2Kmmi

  // stage 4: uuT / Csum, W = cho_solve(L, Csum) for tr_kl
  uut_kernel<<<256, 256, 0, stream>>>(qmu, qs, uuT, Csum);
  trsm_fwd<<<1, 256, 0, stream>>>(Lm, Csum, tmm, Mm);
  trsm_bwd<<<1, 256, 0, stream>>>(Lm, tmm, Wm, Mm);

  // stage 5: forward moments
  fmean_kernel<<<128, 256, 0, stream>>>(PT, qmu, out + FM_OFF);
  h_kernel<<<1024, 256, 0, stream>>>(PT, uuT, H);
  for (int c = 0; c < 8; ++c) {
    int n0c = c * 128;
    tmul_kernel<<<4096, 32, 0, stream>>>(PT, qs, Tc, n0c);
    gram_kernel<<<512, 32, 0, stream>>>(Tc, betap, out + FV_OFF, n0c);
  }

  // stage 6: scalar lml (deterministic single-block reductions)
  lml_kernel<<<1, 256, 0, stream>>>(KP2, Wm, Lm, qs, qmu, alph, KP2K, uuT, H,
                                    PT, varp, betap, out + LML_OFF);
}